// GraphODEFunc_GNODE_7035156431295
// MI455X (gfx1250) — compile-verified
//
#include <hip/hip_runtime.h>
#include <hip/hip_bf16.h>
#include <math.h>

typedef float v8f __attribute__((ext_vector_type(8)));
typedef float v2f __attribute__((ext_vector_type(2)));

#define FDIM 128

// ---------------- degree / normalization ----------------

__global__ void zero_f32(float* __restrict__ p, int n) {
  int i = blockIdx.x * blockDim.x + threadIdx.x;
  if (i < n) p[i] = 0.0f;
}

__global__ void deg_accum(const int* __restrict__ dst, float* __restrict__ deg, int E) {
  int e = blockIdx.x * blockDim.x + threadIdx.x;
  if (e < E) atomicAdd(&deg[dst[e]], 1.0f);
}

__global__ void deg_to_dis(float* __restrict__ deg, int n) {
  int i = blockIdx.x * blockDim.x + threadIdx.x;
  if (i < n) deg[i] = rsqrtf(deg[i] + 1.0f);
}

// ---------------- fp32 WMMA GEMM + self-loop seed ----------------
// hlin = X @ W           (N x 128) @ (128 x 128)
// agg  = hlin * dis^2 + bias   (seed for edge aggregation)
// Block = 256 threads = 8 waves; block b covers rows [16b,16b+16),
// wave w covers cols [16w,16w+16). K-loop: 32 x V_WMMA_F32_16X16X4_F32.
__global__ void gemm16_wmma_seed(const float* __restrict__ X,
                                 const float* __restrict__ W,
                                 const float* __restrict__ bias,
                                 const float* __restrict__ dis,
                                 float* __restrict__ hlin,
                                 float* __restrict__ agg) {
  const int lane = threadIdx.x & 31;
  const int wave = threadIdx.x >> 5;   // 0..7 -> N tile
  const int half = lane >> 4;          // 0: lanes 0-15, 1: lanes 16-31
  const int r    = lane & 15;
  const int m0   = blockIdx.x * 16;
  const int n0   = wave * 16;

  v8f acc = {};
#pragma unroll 4
  for (int k = 0; k < FDIM; k += 4) {
    // A 16x4 f32: lane half 0 holds K=k..k+1, half 1 holds K=k+2..k+3 of row m0+r
    v2f a = *(const v2f*)(X + (size_t)(m0 + r) * FDIM + k + 2 * half);
    // B 4x16 f32: mirrored striping, col n0+r from W rows (k+2*half), (k+2*half+1)
    const float* bp = W + (size_t)(k + 2 * half) * FDIM + n0 + r;
    v2f bm;
    bm.x = bp[0];
    bm.y = bp[FDIM];
    acc = __builtin_amdgcn_wmma_f32_16x16x4_f32(false, a, false, bm,
                                                (short)0, acc, false, false);
  }

  const int   col  = n0 + r;
  const float bcol = bias[col];
#pragma unroll
  for (int v = 0; v < 8; ++v) {
    const int   row = m0 + v + 8 * half;  // C/D layout: VGPR v, half-wave selects M or M+8
    const float hv  = acc[v];
    const float ds  = dis[row];
    hlin[(size_t)row * FDIM + col] = hv;
    agg [(size_t)row * FDIM + col] = hv * ds * ds + bcol;
  }
}

// ---------------- edge scatter (dominant phase, L2-resident) ----------------
// One wave per edge; lane l handles features [4l, 4l+4): float4 gather of the
// source row, 4 global_atomic_add_f32 into the destination row.
__global__ void edge_scatter(const float* __restrict__ hlin,
                             const float* __restrict__ dis,
                             const int* __restrict__ src,
                             const int* __restrict__ dst,
                             float* __restrict__ agg, int E) {
  const int e = blockIdx.x * (blockDim.x >> 5) + (threadIdx.x >> 5);
  if (e >= E) return;
  const int lane = threadIdx.x & 31;
  const int s = src[e];
  const int d = dst[e];
  const float norm = dis[s] * dis[d];
  const float4 v = *(const float4*)(hlin + (size_t)s * FDIM + lane * 4);
  float* out = agg + (size_t)d * FDIM + lane * 4;
  atomicAdd(out + 0, v.x * norm);
  atomicAdd(out + 1, v.y * norm);
  atomicAdd(out + 2, v.z * norm);
  atomicAdd(out + 3, v.w * norm);
}

__global__ void tanh_kernel(const float* __restrict__ in, float* __restrict__ out, int n) {
  int i = blockIdx.x * blockDim.x + threadIdx.x;
  if (i < n) out[i] = tanhf(in[i]);
}

// ---------------- driver ----------------

extern "C" void kernel_launch(void* const* d_in, const int* in_sizes, int n_in,
                              void* d_out, int out_size, void* d_ws, size_t ws_size,
                              hipStream_t stream) {
  // inputs: 0=t, 1=x(N,F), 2=edge_index(2,E), 3=W1, 4=b1, 5=W2, 6=b2, 7=W3, 8=b3
  const float* x  = (const float*)d_in[1];
  const int*   ei = (const int*)  d_in[2];
  const float* W1 = (const float*)d_in[3];
  const float* b1 = (const float*)d_in[4];
  const float* W2 = (const float*)d_in[5];
  const float* b2 = (const float*)d_in[6];
  const float* W3 = (const float*)d_in[7];
  const float* b3 = (const float*)d_in[8];

  const int N = in_sizes[1] / FDIM;   // 10000
  const int E = in_sizes[2] / 2;      // 640000
  const int* src = ei;                // edge_index[0]
  const int* dst = ei + E;            // edge_index[1]

  // workspace layout (float, 16B-aligned slabs)
  const size_t Npad = ((size_t)N + 3) & ~(size_t)3;
  const size_t NH   = (size_t)N * FDIM;
  float* dis  = (float*)d_ws;         // N       (deg -> dis in place)
  float* hlin = dis + Npad;           // N*128   raw X@W (edge messages)
  float* agg  = hlin + NH;            // N*128   aggregation buffer
  float* h    = agg + NH;             // N*128   tanh output / next-layer input
  float* out  = (float*)d_out;

  const int tpb      = 256;
  const int nblk     = (N + tpb - 1) / tpb;
  const int eblk_thr = (E + tpb - 1) / tpb;
  const int mblocks  = N / 16;                 // 625 (N = 625*16 exactly)
  const int eblocks  = (E + 7) / 8;            // 8 edges (waves) per block
  const int nhblocks = (int)((NH + tpb - 1) / tpb);

  // normalization: dis = rsqrt(deg_in + 1)
  zero_f32  <<<nblk,     tpb, 0, stream>>>(dis, N);
  deg_accum <<<eblk_thr, tpb, 0, stream>>>(dst, dis, E);
  deg_to_dis<<<nblk,     tpb, 0, stream>>>(dis, N);

  // layer 1
  gemm16_wmma_seed<<<mblocks, tpb, 0, stream>>>(x, W1, b1, dis, hlin, agg);
  edge_scatter    <<<eblocks, tpb, 0, stream>>>(hlin, dis, src, dst, agg, E);
  tanh_kernel     <<<nhblocks, tpb, 0, stream>>>(agg, h, (int)NH);

  // layer 2
  gemm16_wmma_seed<<<mblocks, tpb, 0, stream>>>(h, W2, b2, dis, hlin, agg);
  edge_scatter    <<<eblocks, tpb, 0, stream>>>(hlin, dis, src, dst, agg, E);
  tanh_kernel     <<<nhblocks, tpb, 0, stream>>>(agg, h, (int)NH);

  // layer 3: seed + aggregate directly into d_out (no tanh)
  gemm16_wmma_seed<<<mblocks, tpb, 0, stream>>>(h, W3, b3, dis, hlin, out);
  edge_scatter    <<<eblocks, tpb, 0, stream>>>(hlin, dis, src, dst, out, E);
}